// Phi3SmallSelfAttention_33715493273943
// MI455X (gfx1250) — compile-verified
//
#include <hip/hip_runtime.h>
#include <cstddef>
#include <cstdint>

typedef __attribute__((ext_vector_type(2))) float v2f;
typedef __attribute__((ext_vector_type(8))) float v8f;

constexpr int S      = 2048;
constexpr int HIDDEN = 4096;
constexpr int NH     = 32;
constexpr int NKV    = 8;
constexpr int HD     = 128;
constexpr int QPK    = NH / NKV;                   // 4
constexpr int QKV_OUT = NKV * (QPK + 2) * HD;      // 6144
constexpr int LOCAL  = 16;
constexpr int VERT   = 8;
constexpr float SCALE = 1.0f / (float)HD;

__device__ __forceinline__ v8f wmma4(v2f a, v2f b, v8f c) {
  // D = A(16x4, f32) * B(4x16, f32) + C(16x16, f32)
  return __builtin_amdgcn_wmma_f32_16x16x4_f32(false, a, false, b, (short)0, c,
                                               false, false);
}

// Async DMA: global -> LDS, 16 bytes per lane, tracked by ASYNCcnt.
__device__ __forceinline__ void async_ld_b128(void* lds, const void* gaddr) {
  asm volatile("global_load_async_to_lds_b128 %0, %1, off"
               :: "v"((unsigned)(uintptr_t)lds), "v"(gaddr)
               : "memory");
}
__device__ __forceinline__ void wait_async0() {
  asm volatile("s_wait_asynccnt 0x0" ::: "memory");
}

// ---------------------------------------------------------------------------
// GEMM (NT):  C[M,N] = A[M,K] @ W[N,K]^T + bias[N]
// grid = (N/128, M/128), block = 256 (8 waves); wave tile = 64x32.
// Double-buffered LDS filled by async-to-LDS DMA (no VGPR staging):
//   issue DMA for chunk i+1, compute chunk i, s_wait_asynccnt 0, barrier.
// ---------------------------------------------------------------------------
constexpr int BM = 128, BN = 128, BK = 16;

__global__ void __launch_bounds__(256) gemm_nt_bias_kernel(
    const float* __restrict__ A, const float* __restrict__ W,
    const float* __restrict__ bias, float* __restrict__ C, int N, int K) {
  __shared__ float As[2][BM][BK + 1];
  __shared__ float Ws[2][BN][BK + 1];

  const int tid  = threadIdx.x;
  const int lane = tid & 31;
  const int wave = tid >> 5;
  const int half = lane >> 4;
  const int l    = lane & 15;

  const int m_blk = blockIdx.y * BM;
  const int n_blk = blockIdx.x * BN;
  const int wm = (wave >> 2) * 64;  // 0 / 64
  const int wn = (wave & 3) * 32;   // 0,32,64,96

  // fill assignment: rows (tid>>2) and 64+(tid>>2), 4 floats at (tid&3)*4
  const int grow = tid >> 2;        // 0..63
  const int gc4  = (tid & 3) << 2;  // 0,4,8,12
  const float* Ap0 = A + (size_t)(m_blk + grow) * K + gc4;
  const float* Ap1 = A + (size_t)(m_blk + 64 + grow) * K + gc4;
  const float* Wp0 = W + (size_t)(n_blk + grow) * K + gc4;
  const float* Wp1 = W + (size_t)(n_blk + 64 + grow) * K + gc4;

  v8f acc[4][2] = {};

  // preload chunk 0 via async DMA
  async_ld_b128(&As[0][grow][gc4],      Ap0);
  async_ld_b128(&As[0][64 + grow][gc4], Ap1);
  async_ld_b128(&Ws[0][grow][gc4],      Wp0);
  async_ld_b128(&Ws[0][64 + grow][gc4], Wp1);
  wait_async0();
  __syncthreads();

  const int nchunks = K / BK;
  for (int i = 0; i < nchunks; ++i) {
    const int cur = i & 1;
    const bool more = (i + 1) < nchunks;
    if (more) {  // uniform branch: DMA next chunk while computing this one
      const int k0  = (i + 1) * BK;
      const int nxt = cur ^ 1;
      async_ld_b128(&As[nxt][grow][gc4],      Ap0 + k0);
      async_ld_b128(&As[nxt][64 + grow][gc4], Ap1 + k0);
      async_ld_b128(&Ws[nxt][grow][gc4],      Wp0 + k0);
      async_ld_b128(&Ws[nxt][64 + grow][gc4], Wp1 + k0);
    }

#pragma unroll
    for (int kk = 0; kk < BK; kk += 4) {
      const int ka = kk + (half << 1);
      v2f af[4], bf[2];
#pragma unroll
      for (int r = 0; r < 4; ++r) {
        af[r].x = As[cur][wm + r * 16 + l][ka];
        af[r].y = As[cur][wm + r * 16 + l][ka + 1];
      }
#pragma unroll
      for (int c = 0; c < 2; ++c) {
        bf[c].x = Ws[cur][wn + c * 16 + l][ka];
        bf[c].y = Ws[cur][wn + c * 16 + l][ka + 1];
      }
#pragma unroll
      for (int r = 0; r < 4; ++r)
#pragma unroll
        for (int c = 0; c < 2; ++c)
          acc[r][c] = wmma4(af[r], bf[c], acc[r][c]);
    }

    wait_async0();   // next chunk's DMA has landed (overlapped with compute)
    __syncthreads(); // publish it; all waves done reading buffer `cur`
  }

#pragma unroll
  for (int r = 0; r < 4; ++r)
#pragma unroll
    for (int c = 0; c < 2; ++c) {
      const int col = n_blk + wn + c * 16 + l;
      const float bv = bias[col];
#pragma unroll
      for (int v = 0; v < 8; ++v) {
        const int row = m_blk + wm + r * 16 + v + (half << 3);
        C[(size_t)row * N + col] = acc[r][c][v] + bv;
      }
    }
}

// ---------------------------------------------------------------------------
// RoPE + repack:  qkv[S, NKV, 6, HD] -> Qw[NH,S,HD], Kw[NKV,S,HD], Vw[NKV,S,HD]
// grid = S, block = 256
// ---------------------------------------------------------------------------
__global__ void __launch_bounds__(256) rope_split_kernel(
    const float* __restrict__ qkv, const int* __restrict__ positions,
    float* __restrict__ Qw, float* __restrict__ Kw, float* __restrict__ Vw) {
  const int s   = blockIdx.x;
  const int tid = threadIdx.x;
  const float pos = (float)positions[s];
  const float* base = qkv + (size_t)s * QKV_OUT;

  const int slot0 = tid >> 6;  // 0..3 (4 rope slots processed concurrently)
  const int p     = tid & 63;  // pair index within head dim
  const float inv_freq = __powf(1.0e6f, -(float)p * (1.0f / 64.0f));
  const float fr = pos * inv_freq;
  float sn, cs;
  __sincosf(fr, &sn, &cs);

  // 40 rope slots: 8 kv * (4 q + 1 k)
#pragma unroll
  for (int g = 0; g < 10; ++g) {
    const int slot = g * 4 + slot0;  // 0..39
    const int kv = slot / 5;
    const int j  = slot % 5;
    const float* src = base + (kv * 6 + j) * HD;
    const float x1 = src[p];
    const float x2 = src[p + 64];
    const float o1 = x1 * cs - x2 * sn;
    const float o2 = x2 * cs + x1 * sn;
    float* dst = (j < QPK) ? (Qw + ((size_t)(kv * QPK + j) * S + s) * HD)
                           : (Kw + ((size_t)kv * S + s) * HD);
    dst[p]      = o1;
    dst[p + 64] = o2;
  }
  // V passthrough
  for (int i = tid; i < NKV * HD; i += 256) {
    const int kv = i >> 7, d = i & 127;
    Vw[((size_t)kv * S + s) * HD + d] = base[(kv * 6 + QPK + 1) * HD + d];
  }
}

// ---------------------------------------------------------------------------
// Blocksparse flash attention.  grid = (S/64, NH), block = 256 (8 waves).
// Block (qb,h) processes key blocks kb where
//   (qb-kb < LOCAL) || ((kb+h+1) % VERT == 0), kb <= qb   (uniform per block)
// ---------------------------------------------------------------------------
constexpr int QB = 64;
constexpr int KB = 64;

__device__ __forceinline__ bool kb_active(int qb, int h, int kb) {
  return ((qb - kb) < LOCAL) || (((kb + h + 1) % VERT) == 0);
}

__global__ void __launch_bounds__(256) attn_kernel(
    const float* __restrict__ Qw, const float* __restrict__ Kw,
    const float* __restrict__ Vw, float* __restrict__ O) {
  __shared__ float Qs[QB][HD + 4];
  __shared__ float Ks[KB][HD + 4];
  __shared__ float Vs[KB][HD + 4];
  __shared__ float Ss[QB][KB + 1];
  __shared__ float mrow[QB], lrow[QB], srow[QB];
  __shared__ float pmax[QB][4], psum[QB][4];

  const int qb  = blockIdx.x;
  const int h   = blockIdx.y;
  const int kvh = h >> 2;  // h / QPK
  const int tid = threadIdx.x;
  const int lane = tid & 31, wave = tid >> 5;
  const int half = lane >> 4, l = lane & 15;
  const int wm = (wave >> 1) * 16;  // query-row offset: 0,16,32,48
  const int wn = (wave & 1);        // 0/1

  // Q block 64x128 -> LDS via async DMA
  {
    const float* Qb = Qw + ((size_t)h * S + (size_t)qb * QB) * HD;
    for (int i = tid; i < QB * HD / 4; i += 256) {
      const int row = i >> 5;
      const int c4  = (i & 31) << 2;
      async_ld_b128(&Qs[row][c4], Qb + row * HD + c4);
    }
  }
  if (tid < QB) { mrow[tid] = -3.0e38f; lrow[tid] = 0.0f; }
  wait_async0();
  __syncthreads();

  v8f oacc[4] = {};

  for (int kb = 0; kb <= qb; ++kb) {
    if (!kb_active(qb, h, kb)) continue;  // uniform -> EXEC stays all-ones

    const float* Kb = Kw + ((size_t)kvh * S + (size_t)kb * KB) * HD;
    const float* Vb = Vw + ((size_t)kvh * S + (size_t)kb * KB) * HD;
    for (int i = tid; i < KB * HD / 4; i += 256) {
      const int row = i >> 5;
      const int c4  = (i & 31) << 2;
      async_ld_b128(&Ks[row][c4], Kb + row * HD + c4);
      async_ld_b128(&Vs[row][c4], Vb + row * HD + c4);
    }
    // prefetch next active K/V block toward L2/WGP$ (global_prefetch_b8)
    {
      int kb2 = kb + 1;
      while (kb2 <= qb && !kb_active(qb, h, kb2)) ++kb2;
      if (kb2 <= qb) {
        const float* Kn = Kw + ((size_t)kvh * S + (size_t)kb2 * KB) * HD;
        const float* Vn = Vw + ((size_t)kvh * S + (size_t)kb2 * KB) * HD;
        __builtin_prefetch(Kn + tid * 32, 0, 0);  // 256 thr * 128B = 32KB
        __builtin_prefetch(Vn + tid * 32, 0, 0);
      }
    }
    wait_async0();
    __syncthreads();

    // scores: this wave computes rows [wm,wm+16) x cols [wn*32, wn*32+32)
    v8f sc[2] = {};
    const int sn = wn * 32;
#pragma unroll
    for (int kd = 0; kd < HD; kd += 4) {
      const int ka = kd + (half << 1);
      v2f a, b0, b1;
      a.x  = Qs[wm + l][ka];       a.y  = Qs[wm + l][ka + 1];
      b0.x = Ks[sn + l][ka];       b0.y = Ks[sn + l][ka + 1];
      b1.x = Ks[sn + 16 + l][ka];  b1.y = Ks[sn + 16 + l][ka + 1];
      sc[0] = wmma4(a, b0, sc[0]);
      sc[1] = wmma4(a, b1, sc[1]);
    }
    const bool diag = (kb == qb);
#pragma unroll
    for (int t = 0; t < 2; ++t) {
      const int col = sn + t * 16 + l;
#pragma unroll
      for (int v = 0; v < 8; ++v) {
        const int row = wm + v + (half << 3);
        float val = sc[t][v] * SCALE;
        if (diag && col > row) val = -3.0e38f;  // causal (positions == arange)
        Ss[row][col] = val;
      }
    }
    __syncthreads();

    // online softmax update: 4 threads per query row
    {
      const int r    = tid >> 2;  // 0..63
      const int part = tid & 3;   // 16-column slice
      const int c0   = part * 16;
      float mx = -3.0e38f;
#pragma unroll
      for (int c = 0; c < 16; ++c) mx = fmaxf(mx, Ss[r][c0 + c]);
      pmax[r][part] = mx;
      __syncthreads();

      const float m_old = mrow[r];
      float mnew = fmaxf(fmaxf(pmax[r][0], pmax[r][1]),
                         fmaxf(pmax[r][2], pmax[r][3]));
      mnew = fmaxf(mnew, m_old);
      float sum = 0.0f;
#pragma unroll
      for (int c = 0; c < 16; ++c) {
        const float pv = __expf(Ss[r][c0 + c] - mnew);
        Ss[r][c0 + c] = pv;
        sum += pv;
      }
      psum[r][part] = sum;
      __syncthreads();

      if (part == 0) {
        const float rsc = __expf(m_old - mnew);
        mrow[r] = mnew;
        lrow[r] = lrow[r] * rsc + psum[r][0] + psum[r][1] + psum[r][2] + psum[r][3];
        srow[r] = rsc;
      }
    }
    __syncthreads();

    // rescale accumulators, then oacc += P @ V
#pragma unroll
    for (int j = 0; j < 4; ++j)
#pragma unroll
      for (int v = 0; v < 8; ++v)
        oacc[j][v] *= srow[wm + v + (half << 3)];

#pragma unroll
    for (int kk = 0; kk < KB; kk += 4) {
      const int ka = kk + (half << 1);
      v2f a;
      a.x = Ss[wm + l][ka]; a.y = Ss[wm + l][ka + 1];
#pragma unroll
      for (int j = 0; j < 4; ++j) {
        const int nc = wn * 64 + j * 16;
        v2f b;
        b.x = Vs[ka][nc + l]; b.y = Vs[ka + 1][nc + l];
        oacc[j] = wmma4(a, b, oacc[j]);
      }
    }
    __syncthreads();
  }

  // O[(qb*64+row), h*HD + col], row-major [S, NH*HD]
#pragma unroll
  for (int j = 0; j < 4; ++j) {
    const int col = wn * 64 + j * 16 + l;
#pragma unroll
    for (int v = 0; v < 8; ++v) {
      const int row = wm + v + (half << 3);
      O[(size_t)(qb * QB + row) * (NH * HD) + (size_t)h * HD + col] =
          oacc[j][v] / lrow[row];
    }
  }
}

// ---------------------------------------------------------------------------
extern "C" void kernel_launch(void* const* d_in, const int* in_sizes, int n_in,
                              void* d_out, int out_size, void* d_ws,
                              size_t ws_size, hipStream_t stream) {
  (void)in_sizes; (void)n_in; (void)out_size; (void)ws_size;
  const int*   positions = (const int*)d_in[0];
  const float* hidden    = (const float*)d_in[1];
  const float* w_qkv     = (const float*)d_in[2];
  const float* b_qkv     = (const float*)d_in[3];
  const float* w_dense   = (const float*)d_in[4];
  const float* b_dense   = (const float*)d_in[5];
  float* out = (float*)d_out;

  float* ws      = (float*)d_ws;
  float* qkv_raw = ws;                                  // S*QKV_OUT floats
  float* Qw      = qkv_raw + (size_t)S * QKV_OUT;       // NH*S*HD
  float* Kw      = Qw + (size_t)NH * S * HD;            // NKV*S*HD
  float* Vw      = Kw + (size_t)NKV * S * HD;           // NKV*S*HD
  float* attn    = qkv_raw;  // reuse: qkv_raw dead after rope; S*NH*HD fits

  const dim3 blk(256);
  gemm_nt_bias_kernel<<<dim3(QKV_OUT / BN, S / BM), blk, 0, stream>>>(
      hidden, w_qkv, b_qkv, qkv_raw, QKV_OUT, HIDDEN);
  rope_split_kernel<<<dim3(S), blk, 0, stream>>>(qkv_raw, positions, Qw, Kw, Vw);
  attn_kernel<<<dim3(S / QB, NH), blk, 0, stream>>>(Qw, Kw, Vw, attn);
  gemm_nt_bias_kernel<<<dim3(HIDDEN / BN, S / BM), blk, 0, stream>>>(
      attn, w_dense, b_dense, out, HIDDEN, HIDDEN);
}